// Iterative_Layer_1_VertexUpdate_91096256348956
// MI455X (gfx1250) — compile-verified
//
#include <hip/hip_runtime.h>

// Vector types that __builtin_nontemporal_load accepts directly.
typedef int   __attribute__((ext_vector_type(4))) vi4;
typedef float __attribute__((ext_vector_type(4))) vf4;
typedef float __attribute__((ext_vector_type(2))) vf2;

// Global-addrspace float, to guarantee global_atomic_* (not flat_atomic_*).
typedef __attribute__((address_space(1))) float gfloat;

// ---------------------------------------------------------------------------
// Hot kernel: branchless 4-edges-per-lane scatter-add. Every thread with
// t < nQuads is fully in bounds by construction (nQuads = E/4 floor), so the
// body is a straight line: clamped prefetch ~32KB ahead, two NT b128 loads,
// four no-return relaxed agent-scope f32 atomics.
// ---------------------------------------------------------------------------
__global__ void scatter4_kernel(const int* __restrict__ src_idx,
                                const float* __restrict__ edge_w,
                                float* __restrict__ out,
                                int nQuads, int nE) {
    const int t = blockIdx.x * blockDim.x + threadIdx.x;
    if (t < nQuads) {
        const int base = t * 4;

        // Branchless clamp of the prefetch target to the last valid quad.
        int pf = base + 8192;
        int lim = nE - 4;
        pf = pf < lim ? pf : lim;
        __builtin_prefetch(src_idx + pf, 0, 0);
        __builtin_prefetch(edge_w + pf, 0, 0);

        vi4 idx = __builtin_nontemporal_load((const vi4*)(src_idx + base));
        vf4 w   = __builtin_nontemporal_load((const vf4*)(edge_w + base));

        gfloat* gout = (gfloat*)out;
#pragma unroll
        for (int k = 0; k < 4; ++k) {
            gfloat* p = gout + ((long)idx[k] << 1);  // column 0 of row idx[k]
            (void)__hip_atomic_fetch_add(p, w[k], __ATOMIC_RELAXED,
                                         __HIP_MEMORY_SCOPE_AGENT);
        }
    }
}

// ---------------------------------------------------------------------------
// Tail kernel: handles the E % 4 leftover edges (at most 3), one per thread.
// ---------------------------------------------------------------------------
__global__ void scatter_tail_kernel(const int* __restrict__ src_idx,
                                    const float* __restrict__ edge_w,
                                    float* __restrict__ out,
                                    int start, int nE) {
    const int e = start + blockIdx.x * blockDim.x + threadIdx.x;
    if (e < nE) {
        gfloat* p = (gfloat*)out + ((long)src_idx[e] << 1);
        (void)__hip_atomic_fetch_add(p, edge_w[e], __ATOMIC_RELAXED,
                                     __HIP_MEMORY_SCOPE_AGENT);
    }
}

// ---------------------------------------------------------------------------
// Init kernel: out[i] = { 0.0f (cbar accumulator), vertex_attr[i].y }
// Vectorized 8B load + 8B store; also clears the atomic target column
// (harness poisons d_out with 0xAA).
// ---------------------------------------------------------------------------
__global__ void init_out_kernel(const float* __restrict__ vattr,
                                float* __restrict__ out, int V) {
    int i = blockIdx.x * blockDim.x + threadIdx.x;
    if (i < V) {
        vf2 va = *(const vf2*)(vattr + 2 * (long)i);
        vf2 o;
        o.x = 0.0f;   // cbar_i accumulator
        o.y = va.y;   // y_i passthrough
        *(vf2*)(out + 2 * (long)i) = o;
    }
}

// ---------------------------------------------------------------------------
// Launch: inputs are [vertex_attr (V*2 f32), edgeij_pair (2*E i32),
//                     edge_attr (E f32), g (1 f32), batch (V i32)]
// Output: (V, 2) f32 = concat([cbar, y], axis=1).
// ---------------------------------------------------------------------------
extern "C" void kernel_launch(void* const* d_in, const int* in_sizes, int n_in,
                              void* d_out, int out_size, void* d_ws, size_t ws_size,
                              hipStream_t stream) {
    const float* vattr = (const float*)d_in[0];
    const int*   eij   = (const int*)d_in[1];   // row-major (2, E): row 0 first
    const float* ew    = (const float*)d_in[2];
    float*       out   = (float*)d_out;

    const int V = in_sizes[0] / 2;
    const int E = in_sizes[2];
    const int* src = eij;  // edgeij_pair[0] = first E entries

    // Phase 1: initialize output (zero accumulator column + copy y column).
    {
        const int threads = 256;  // 8 wave32 waves per block
        const int blocks  = (V + threads - 1) / threads;
        init_out_kernel<<<blocks, threads, 0, stream>>>(vattr, out, V);
    }

    // Phase 2: atomic scatter-add, 4 edges per lane, all in bounds.
    const int nQuads = E / 4;
    if (nQuads > 0) {
        const int threads = 256;
        const int blocks  = (nQuads + threads - 1) / threads;
        scatter4_kernel<<<blocks, threads, 0, stream>>>(src, ew, out, nQuads, E);
    }

    // Phase 3: leftover E % 4 edges (none for E = 32M, kept for generality).
    const int rem = E - nQuads * 4;
    if (rem > 0) {
        scatter_tail_kernel<<<1, 32, 0, stream>>>(src, ew, out, nQuads * 4, E);
    }
}